// DynamicRouting_67723044323987
// MI455X (gfx1250) — compile-verified
//
#include <hip/hip_runtime.h>

typedef __attribute__((ext_vector_type(2))) float v2f;
typedef __attribute__((ext_vector_type(8))) float v8f;

#define G        8
#define FI       32
#define FO       16
#define CIN      (G * FI)      /* 256 input channels */
#define KCHUNKS  (FI / 4)      /* 8 WMMA K-steps of 4 */
#define NWAVES   8             /* waves per block (256 threads, wave32) */
#define AROW     260           /* padded LDS row stride (floats): banks 4m+c, conflict-free */

__global__ __launch_bounds__(256)
void DynamicRouting_67723044323987_kernel(const float* __restrict__ x,
                                          const float* __restrict__ kern,
                                          const float* __restrict__ bias,
                                          const int*   __restrict__ n_iter_p,
                                          float*       __restrict__ out,
                                          int tiles)
{
    // B fragments packed per lane: Kpk[((g*KCHUNKS+kc)*32 + lane)*2 + {0,1}]
    __shared__ alignas(16) float Kpk[G * KCHUNKS * 32 * 2];   // 4096 floats = 16 KB
    __shared__ alignas(16) float Ash[NWAVES * 16 * AROW];     // ~130 KB x-tile staging

    const int tid = threadIdx.x;
    // pack kernels [g][k][n] -> WMMA B-fragment order (one v2f per lane per (g,kc))
    for (int i = tid; i < G * KCHUNKS * 64; i += 256) {
        const int v  = i & 1;
        const int ln = (i >> 1) & 31;
        const int kc = (i >> 6) & 7;
        const int g  = i >> 9;
        const int hf = ln >> 4, n = ln & 15;
        Kpk[i] = kern[g * (FI * FO) + (kc * 4 + hf * 2 + v) * FO + n];
    }
    __syncthreads();

    const int wave = tid >> 5;
    const int lane = tid & 31;
    const int half = lane >> 4;      // which 16-lane half of the wave32
    const int m    = lane & 15;      // A row (pixel-in-tile) and B/C column (fo)
    const int niter = n_iter_p[0];
    const float bval = bias[m];

    float* Aw = &Ash[wave * 16 * AROW];
    const unsigned abase = (unsigned)(size_t)(void*)Aw;   // low 32 bits = LDS byte offset

    for (int tile = blockIdx.x * NWAVES + wave; tile < tiles;
         tile += gridDim.x * NWAVES) {
        const long pix0 = (long)tile * 16;
        const float* gbase = x + pix0 * CIN;

        // ---- stage 16x256 f32 x-tile into LDS: coalesced async b128 (ASYNCcnt) ----
        for (int row = 0; row < 16; ++row) {
            #pragma unroll
            for (int h = 0; h < 2; ++h) {
                const int col = h * 128 + lane * 4;                 // 16B per lane
                const unsigned laddr = abase + (unsigned)((row * AROW + col) * 4);
                const unsigned long long gaddr =
                    (unsigned long long)(size_t)(gbase + row * CIN + col);
                asm volatile("global_load_async_to_lds_b128 %0, %1, off"
                             :: "v"(laddr), "v"(gaddr) : "memory");
            }
        }
        asm volatile("s_wait_asynccnt 0x0" ::: "memory");

        // ---- con[16 pix, g, 16 fo] via f32 WMMA ----
        v8f czero = {};
        v8f cacc[G];
        #pragma unroll
        for (int g = 0; g < G; ++g) cacc[g] = czero;

        const float* Arow = Aw + m * AROW + half * 2;
        #pragma unroll 2
        for (int kc = 0; kc < KCHUNKS; ++kc) {          // runtime loop: bounds staging regs
            #pragma unroll
            for (int g = 0; g < G; ++g) {
                // A 16x4: lanes 0-15 -> K={4kc,4kc+1}, lanes 16-31 -> K={4kc+2,4kc+3}
                v2f a = *(const v2f*)(Arow + g * FI + kc * 4);
                // B 4x16: same K striping, pre-packed per lane
                v2f b = *(const v2f*)&Kpk[((g * KCHUNKS + kc) * 32 + lane) * 2];
                cacc[g] = __builtin_amdgcn_wmma_f32_16x16x4_f32(
                    /*neg_a=*/false, a, /*neg_b=*/false, b,
                    /*c_mod=*/(short)0, cacc[g],
                    /*reuse_a=*/false, /*reuse_b=*/false);
            }
        }

        // ---- dynamic routing (per lane: 8 pixels r=0..7, one fo = m) ----
        float beta[G][8];
        #pragma unroll
        for (int g = 0; g < G; ++g)
            #pragma unroll
            for (int r = 0; r < 8; ++r) beta[g][r] = 0.0f;

        float v[8];
        for (int it = 0; ; ++it) {
            #pragma unroll
            for (int r = 0; r < 8; ++r) v[r] = 0.0f;
            #pragma unroll
            for (int g = 0; g < G; ++g) {
                #pragma unroll
                for (int r = 0; r < 8; ++r) {
                    float alpha = 1.0f / (1.0f + __expf(-beta[g][r]));
                    v[r] += alpha * cacc[g][r];
                }
            }
            if (it >= niter - 1) break;
            // beta[pix][g] += sum_fo con*v : butterfly over the 16 lanes of this half
            #pragma unroll
            for (int g = 0; g < G; ++g) {
                #pragma unroll
                for (int r = 0; r < 8; ++r) {
                    float t = cacc[g][r] * v[r];
                    t += __shfl_xor(t, 1);
                    t += __shfl_xor(t, 2);
                    t += __shfl_xor(t, 4);
                    t += __shfl_xor(t, 8);
                    beta[g][r] += t;
                }
            }
        }

        // ---- store: pixel = pix0 + r + 8*half, column = m ----
        float* orow = out + (pix0 + half * 8) * FO + m;
        #pragma unroll
        for (int r = 0; r < 8; ++r)
            orow[r * FO] = v[r] + bval;
    }
}

extern "C" void kernel_launch(void* const* d_in, const int* in_sizes, int n_in,
                              void* d_out, int out_size, void* d_ws, size_t ws_size,
                              hipStream_t stream) {
    (void)n_in; (void)out_size; (void)d_ws; (void)ws_size;
    const float* x      = (const float*)d_in[0];
    const float* kern   = (const float*)d_in[1];
    const float* bias   = (const float*)d_in[2];
    const int*   n_iter = (const int*)d_in[3];
    float*       out    = (float*)d_out;

    const int P     = in_sizes[0] / CIN;  // 131072 pixels
    const int tiles = P / 16;             // 8192 16-pixel tiles
    int blocks = (tiles + NWAVES - 1) / NWAVES;   // 1024
    if (blocks < 1) blocks = 1;

    DynamicRouting_67723044323987_kernel<<<blocks, 256, 0, stream>>>(
        x, kern, bias, n_iter, out, tiles);
}